// PyramidSampler_51694226375009
// MI455X (gfx1250) — compile-verified
//
#include <hip/hip_runtime.h>
#include <stdint.h>

#define B   4
#define P   8192
#define D   512
#define N1  4096
#define N2  2048
#define RPB (P + N1 + N2)   /* 14336 rows per batch in fused output */
#define ROWS (B * RPB)      /* 57344 */

typedef __attribute__((ext_vector_type(2))) float        v2f;
typedef __attribute__((ext_vector_type(8))) float        v8f;
typedef __attribute__((ext_vector_type(4))) unsigned int v4u;
typedef __attribute__((ext_vector_type(8))) int          v8i;
typedef __attribute__((ext_vector_type(4))) int          v4i;

// ---------------------------------------------------------------------------
// Kernel 1: per-point squared norms via V_WMMA_F32_16X16X4_F32.
// A = B = 16x4 tile [x,y,z,0]; C = P.P^T Gram tile; diag(C) = ||p||^2.
// One wave handles one 16-point tile. 2048 tiles total for B*P points.
// ---------------------------------------------------------------------------
__global__ __launch_bounds__(256) void pp_wmma_kernel(const float* __restrict__ pos,
                                                      float* __restrict__ pp) {
  const int lane = threadIdx.x & 31;
  const int wave = threadIdx.x >> 5;
  const int tile = blockIdx.x * 8 + wave;
  const int base = tile * 16;
  const int m    = lane & 15;   // point within tile (lanes 0-15 and 16-31 mirror)
  const int half = lane >> 4;

  const float* p = pos + (size_t)(base + m) * 3;
  const float x_ = p[0], y_ = p[1], z_ = p[2];

  // A (16x4 f32, 2 VGPRs): lanes 0-15 hold K=0,1 ; lanes 16-31 hold K=2,3.
  // B (4x16 f32) uses the mirrored layout, so for P.P^T the operand is identical.
  v2f a;
  a.x = half ? z_   : x_;   // K=0 / K=2
  a.y = half ? 0.0f : y_;   // K=1 / K=3 (pad)

  v8f c = {};
  c = __builtin_amdgcn_wmma_f32_16x16x4_f32(false, a, false, a, (short)0, c,
                                            false, false);

  // diag extraction: point n<8 -> lane n, VGPR n ; point n>=8 -> lane n+16, VGPR n-8
  float outv = 0.0f;
#pragma unroll
  for (int k = 0; k < 8; ++k) {
    if (lane == k)      outv = c[k];
    if (lane == k + 24) outv = c[k];
  }
  if (lane < 8)        pp[base + lane]      = outv;
  else if (lane >= 24) pp[base + lane - 16] = outv;
}

// ---------------------------------------------------------------------------
// Kernel 2: farthest-point sampling. One workgroup per (batch, scale).
// pos tile staged to LDS by the Tensor Data Mover; dist kept in registers
// (8 points/thread); argmax = wave32 shfl reduce + 32-entry LDS reduce.
// dist update uses d = ||p||^2 + ||c||^2 - 2 p.c with WMMA-precomputed ||p||^2.
// ---------------------------------------------------------------------------
__global__ __launch_bounds__(1024) void fps_kernel(const float* __restrict__ pos,
                                                   const float* __restrict__ pp,
                                                   int* __restrict__ idx1,
                                                   int* __restrict__ idx2) {
  __shared__ float spos[P * 3];   // 96 KB, fits CDNA5 320 KB WGP LDS
  __shared__ float rval[32];
  __shared__ int   ridx[32];
  __shared__ int   sfar;

  const int b  = blockIdx.x >> 1;
  const int sc = blockIdx.x & 1;
  const int n  = sc ? N2 : N1;
  int* outidx  = sc ? (idx2 + b * N2) : (idx1 + b * N1);

  const int tid  = threadIdx.x;
  const int lane = tid & 31;
  const int wid  = tid >> 5;

  const float* gpos = pos + (size_t)b * P * 3;

#if defined(__gfx1250__) && __has_builtin(__builtin_amdgcn_tensor_load_to_lds)
  if (tid == 0) {
    const unsigned lds = (unsigned)(uintptr_t)&spos[0];
    const unsigned long long ga = (unsigned long long)(uintptr_t)gpos;
    const unsigned elems = P * 3;                 // 24576 dwords
    v4u g0; v8i g1;
    v4i g2 = {0, 0, 0, 0}, g3 = {0, 0, 0, 0};
    v8i g4 = {0, 0, 0, 0, 0, 0, 0, 0};
    // D# group 0: count=1, lds_addr, global_addr[56:0], type=2
    g0[0] = 1u;
    g0[1] = lds;
    g0[2] = (unsigned)(ga & 0xffffffffu);
    g0[3] = (unsigned)((ga >> 32) & 0x1ffffffu) | (2u << 30);
    // D# group 1: workgroup_mask=0 (not in cluster), data_size=4B,
    // tensor_dim0=24576, tensor_dim1=1, tile_dim0=24576 (1-D tile), stride=24576
    g1[0] = (int)(2u << 16);                         // data_size=2 -> 4 bytes
    g1[1] = (int)((elems & 0xffffu) << 16);          // tensor_dim0[15:0] @bit48
    g1[2] = (int)(((elems >> 16) & 0xffffu) | (1u << 16)); // dim0 hi | dim1 lo=1
    g1[3] = (int)((elems & 0xffffu) << 16);          // dim1 hi=0 | tile_dim0
    g1[4] = 0;                                       // tile_dim1=0, tile_dim2=0
    g1[5] = (int)elems;                              // tensor_dim0_stride lo
    g1[6] = 0;
    g1[7] = 0;
    // 6-arg form (clang-23 / therock-10.0 headers): extra int32x8 group, zeroed
    // for a 1-D tile (dims >= 2 unused), then cache-policy immediate.
    __builtin_amdgcn_tensor_load_to_lds(g0, g1, g2, g3, g4, 0);
#if __has_builtin(__builtin_amdgcn_s_wait_tensorcnt)
    __builtin_amdgcn_s_wait_tensorcnt(0);
#endif
  }
#else
  for (int i = tid; i < P * 3; i += 1024) spos[i] = gpos[i];
#endif
  __syncthreads();

  // Each thread owns 8 points: j = s*1024 + tid (coalesced, conflict-free).
  float ox[8], oy[8], oz[8], opp[8], dl[8];
#pragma unroll
  for (int s = 0; s < 8; ++s) {
    const int j = s * 1024 + tid;
    ox[s]  = spos[j * 3 + 0];
    oy[s]  = spos[j * 3 + 1];
    oz[s]  = spos[j * 3 + 2];
    opp[s] = pp[(size_t)b * P + j];
    dl[s]  = 1e10f;   // BIG
  }

  int far = 0;
  for (int t = 0; t < n; ++t) {
    if (tid == 0) outidx[t] = far;          // record BEFORE update (matches scan)

    const float cx = spos[far * 3 + 0];     // LDS broadcast read
    const float cy = spos[far * 3 + 1];
    const float cz = spos[far * 3 + 2];
    const float cc = cx * cx + cy * cy + cz * cz;

    float bv = -1e30f; int bi = 0x7fffffff;
#pragma unroll
    for (int s = 0; s < 8; ++s) {
      const float d = opp[s] + cc - 2.0f * (ox[s] * cx + oy[s] * cy + oz[s] * cz);
      dl[s] = fminf(dl[s], d);
      const int j = s * 1024 + tid;
      if (dl[s] > bv || (dl[s] == bv && j < bi)) { bv = dl[s]; bi = j; }
    }
    // wave32 argmax (lowest index wins ties, like jnp.argmax)
#pragma unroll
    for (int off = 16; off > 0; off >>= 1) {
      const float ov = __shfl_xor(bv, off, 32);
      const int   oi = __shfl_xor(bi, off, 32);
      if (ov > bv || (ov == bv && oi < bi)) { bv = ov; bi = oi; }
    }
    if (lane == 0) { rval[wid] = bv; ridx[wid] = bi; }
    __syncthreads();
    if (wid == 0) {
      bv = rval[lane]; bi = ridx[lane];
#pragma unroll
      for (int off = 16; off > 0; off >>= 1) {
        const float ov = __shfl_xor(bv, off, 32);
        const int   oi = __shfl_xor(bi, off, 32);
        if (ov > bv || (ov == bv && oi < bi)) { bv = ov; bi = oi; }
      }
      if (lane == 0) sfar = bi;
    }
    __syncthreads();
    far = sfar;
  }
}

// ---------------------------------------------------------------------------
// Kernel 3: fused gather (bandwidth-bound, ~230 MB @ 23.3 TB/s ~ 10 us).
// One block per output row; 128 lanes x float4 = 512 floats.
// ---------------------------------------------------------------------------
__global__ __launch_bounds__(128) void gather_kernel(const float* __restrict__ x,
                                                     const float* __restrict__ pos,
                                                     const int* __restrict__ idx1,
                                                     const int* __restrict__ idx2,
                                                     float* __restrict__ outx,
                                                     float* __restrict__ outp) {
  const int row = blockIdx.x;
  const int b   = row / RPB;
  const int o   = row - b * RPB;
  int src;
  if (o < P)           src = o;                              // scale 1.0: identity
  else if (o < P + N1) src = idx1[b * N1 + (o - P)];         // scale 0.5
  else                 src = idx2[b * N2 + (o - P - N1)];    // scale 0.25
  const int srow = b * P + src;

  const float4* xs = (const float4*)(x + (size_t)srow * D);
  float4*       xd = (float4*)(outx + (size_t)row * D);
  __builtin_prefetch(xs + threadIdx.x, 0, 0);   // global_prefetch_b8
  xd[threadIdx.x] = xs[threadIdx.x];
  if (threadIdx.x < 3)
    outp[(size_t)row * 3 + threadIdx.x] = pos[(size_t)srow * 3 + threadIdx.x];
}

// ---------------------------------------------------------------------------
extern "C" void kernel_launch(void* const* d_in, const int* in_sizes, int n_in,
                              void* d_out, int out_size, void* d_ws, size_t ws_size,
                              hipStream_t stream) {
  (void)in_sizes; (void)n_in; (void)out_size; (void)ws_size;

  const float* x   = (const float*)d_in[0];   // [B*P, D] f32
  const float* pos = (const float*)d_in[1];   // [B*P, 3] f32
  // d_in[2] = batch_idx (int64): sorted equal-sized batches -> layout known

  float* outx = (float*)d_out;                       // fused_x: ROWS * D
  float* outp = outx + (size_t)ROWS * D;             // fused_p: ROWS * 3

  float* pp  = (float*)d_ws;                                        // B*P floats
  int*   idx1 = (int*)((char*)d_ws + sizeof(float) * (size_t)B * P); // B*N1 ints
  int*   idx2 = idx1 + (size_t)B * N1;                               // B*N2 ints

  pp_wmma_kernel<<<(B * P) / (16 * 8), 256, 0, stream>>>(pos, pp);
  fps_kernel<<<B * 2, 1024, 0, stream>>>(pos, pp, idx1, idx2);
  gather_kernel<<<ROWS, 128, 0, stream>>>(x, pos, idx1, idx2, outx, outp);
}